// ShiftedPearsonCorrelation_83829171683480
// MI455X (gfx1250) — compile-verified
//
#include <hip/hip_runtime.h>

typedef float v2f __attribute__((ext_vector_type(2)));
typedef float v8f __attribute__((ext_vector_type(8)));

#define MAXSHIFT 128
#define NSHIFT   257      // 2*128+1
#define CH_STRIDE 1024    // floats of workspace per channel
#define WIN_X    0        // 257 floats
#define WIN_X2   320      // 257 floats
#define CORR     640      // 257 floats
#define MEANY    900
#define YC2      901

#define UCHUNK   4096     // K-chunk staged in LDS per super-iteration
#define YOFF     240      // left padding of the staged y window

// ---------------------------------------------------------------------------
// Kernel 1: per-channel statistics.
// ---------------------------------------------------------------------------
__global__ void stats_kernel(const float* __restrict__ x,
                             const float* __restrict__ y,
                             float* __restrict__ ws, int T) {
  const int bc = blockIdx.x;
  const float* xc = x + (size_t)bc * T;
  const float* yv = y + (size_t)bc * T;
  float* w = ws + (size_t)bc * CH_STRIDE;

  __shared__ float red[4][256];
  float sx = 0.f, sx2 = 0.f, sy = 0.f, sy2 = 0.f;
  for (int t = threadIdx.x; t < T; t += blockDim.x) {
    float xv = xc[t], yvv = yv[t];
    sx += xv; sx2 += xv * xv; sy += yvv; sy2 += yvv * yvv;
  }
  red[0][threadIdx.x] = sx;  red[1][threadIdx.x] = sx2;
  red[2][threadIdx.x] = sy;  red[3][threadIdx.x] = sy2;
  __syncthreads();
  for (int off = 128; off > 0; off >>= 1) {
    if ((int)threadIdx.x < off) {
      red[0][threadIdx.x] += red[0][threadIdx.x + off];
      red[1][threadIdx.x] += red[1][threadIdx.x + off];
      red[2][threadIdx.x] += red[2][threadIdx.x + off];
      red[3][threadIdx.x] += red[3][threadIdx.x + off];
    }
    __syncthreads();
  }
  const float tot_x  = red[0][0], tot_x2 = red[1][0];
  const float tot_y  = red[2][0], tot_y2 = red[3][0];

  const int s = threadIdx.x;
  if (s < NSHIFT) {
    float hx = 0.f, hx2 = 0.f;
    for (int j = 0; j < s - MAXSHIFT; ++j) {       // missing head (s > 128)
      float v = xc[j]; hx += v; hx2 += v * v;
    }
    float tx = 0.f, tx2 = 0.f;
    for (int j = s + T - MAXSHIFT; j < T; ++j) {   // missing tail (s < 128)
      float v = xc[j]; tx += v; tx2 += v * v;
    }
    w[WIN_X  + s] = tot_x  - hx  - tx;
    w[WIN_X2 + s] = tot_x2 - hx2 - tx2;
  }
  if (threadIdx.x == 0) {
    w[MEANY] = tot_y / (float)T;
    w[YC2]   = tot_y2 - tot_y * tot_y / (float)T;
  }
}

// ---------------------------------------------------------------------------
// Kernel 2: WMMA cross-correlation numerator.
//  Shift s = 16*n + m.  D[m,n] = sum_u A[m,u] * B[u,n] with
//    A[m,u] = x_pad[u + m]      (Hankel slice of padded x)
//    B[u,n] = yc_pad[u - 16*n]  (Toeplitz slice of centered, padded y)
//  Staged through LDS in UCHUNK pieces; interior chunks use an unchecked
//  float4 fast path.  y-stage swizzle j -> j + (j>>5) is conflict-free for
//  the -16-lane-stride B reads, keeps b.x/b.y adjacent, and advances by a
//  constant (+33) per K-step.  Inner loop is software-pipelined.
// ---------------------------------------------------------------------------
__device__ __forceinline__ float xpad(const float* __restrict__ xc, int i, int T) {
  int j = i - MAXSHIFT;
  return (j >= 0 && j < T) ? xc[j] : 0.0f;
}
__device__ __forceinline__ float ypad(const float* __restrict__ yv, int i, int T, float mean) {
  return (i >= 0 && i < T) ? (yv[i] - mean) : 0.0f;
}

__global__ void corr_wmma_kernel(const float* __restrict__ x,
                                 const float* __restrict__ y,
                                 float* __restrict__ ws, int T) {
  const int bc = blockIdx.x;
  const float* xc = x + (size_t)bc * T;
  const float* yv = y + (size_t)bc * T;
  float* w = ws + (size_t)bc * CH_STRIDE;
  const float mean = w[MEANY];

  const int lane = threadIdx.x & 31;
  const int wave = threadIdx.x >> 5;   // 0..7
  const int m    = lane & 15;          // A row (M) / B column (N) index
  const int koff = (lane >> 4) * 2;    // lanes 0-15: K=0,1; lanes 16-31: K=2,3

  __shared__ __align__(16) float lds_x[UCHUNK + 32];
  __shared__ float lds_y[UCHUNK + 256 + ((UCHUNK + 256) >> 5) + 8];

  v8f c = {};
  const int Ulen = T + 16 * 15;        // K extent: u in [0, T+240)

  for (int u_base = 0; u_base < Ulen; u_base += UCHUNK) {
    // ---- stage x chunk: xp[u_base .. u_base+UCHUNK+19] ----
    const int NX = UCHUNK + 20;                    // divisible by 4
    if (u_base >= MAXSHIFT && u_base + NX <= T + MAXSHIFT) {
      const float* src = xc + (u_base - MAXSHIFT); // 16B-aligned
      for (int i = threadIdx.x * 4; i < NX; i += 1024) {
        float4 v = *(const float4*)(src + i);
        *(float4*)&lds_x[i] = v;                   // ds_store_b128
      }
    } else {
      for (int i = threadIdx.x; i < NX; i += 256)
        lds_x[i] = xpad(xc, u_base + i, T);
    }
    // ---- stage y chunk: ycp[u_base-240 .. u_base+UCHUNK+3], swizzled ----
    const int NY = UCHUNK + YOFF + 4;              // divisible by 4
    if (u_base >= YOFF && u_base + UCHUNK + 4 <= T) {
      const float* src = yv + (u_base - YOFF);     // 64B-aligned
      for (int i = threadIdx.x * 4; i < NY; i += 1024) {
        float4 v = *(const float4*)(src + i);
        const int sw = i + (i >> 5);               // i..i+3 stay in one 32-block
        lds_y[sw]     = v.x - mean;
        lds_y[sw + 1] = v.y - mean;
        lds_y[sw + 2] = v.z - mean;
        lds_y[sw + 3] = v.w - mean;
      }
    } else {
      for (int i = threadIdx.x; i < NY; i += 256)
        lds_y[i + (i >> 5)] = ypad(yv, u_base - YOFF + i, T, mean);
    }
    __syncthreads();

    // ---- software-pipelined WMMA inner loop ----
    const int chunk_len = ((u_base + UCHUNK < Ulen) ? UCHUNK : (Ulen - u_base));
    const int start = wave * 4;
    const int iters = (chunk_len - start + 31) >> 5;   // >= 1 for all waves

    int ai = start + koff + m;                 // lds_x index, += 32
    int bj = start + koff - 16 * m + YOFF;     // += 32
    int bs = bj + (bj >> 5);                   // swizzled, += 33

    v2f a, b;
    a.x = lds_x[ai];     a.y = lds_x[ai + 1];
    b.x = lds_y[bs];     b.y = lds_y[bs + 1];
    for (int it = 0; it < iters - 1; ++it) {
      ai += 32; bs += 33;
      v2f an, bn;
      an.x = lds_x[ai];  an.y = lds_x[ai + 1];
      bn.x = lds_y[bs];  bn.y = lds_y[bs + 1];
      // (neg_a, A, neg_b, B, c_mod, C, reuse_a, reuse_b)
      c = __builtin_amdgcn_wmma_f32_16x16x4_f32(false, a, false, b,
                                                (short)0, c, false, false);
      a = an; b = bn;
    }
    c = __builtin_amdgcn_wmma_f32_16x16x4_f32(false, a, false, b,
                                              (short)0, c, false, false);
    __syncthreads();
  }

  // ---- reduce the 8 wave partials through LDS ----
  __shared__ float acc[8][32][8];
#pragma unroll
  for (int r = 0; r < 8; ++r) acc[wave][lane][r] = c[r];
  __syncthreads();

  // C/D layout (ISA 7.12.2): VGPR r, lane L -> M = r + 8*(L>=16), N = L&15,
  // shift s = 16*N + M.
  {
    const int L = threadIdx.x & 31;
    const int r = threadIdx.x >> 5;
    float sum = 0.f;
#pragma unroll
    for (int wv = 0; wv < 8; ++wv) sum += acc[wv][L][r];
    const int s = 16 * (L & 15) + r + 8 * (L >> 4);
    w[CORR + s] = sum;
  }
  __syncthreads();

  // ---- final shift s = 256: sum_{t=0}^{T-129} x[t+128] * yc[t] ----
  __shared__ float red[256];
  float p = 0.f;
  for (int t = threadIdx.x; t < T - MAXSHIFT; t += blockDim.x)
    p += xc[t + MAXSHIFT] * (yv[t] - mean);
  red[threadIdx.x] = p;
  __syncthreads();
  for (int off = 128; off > 0; off >>= 1) {
    if ((int)threadIdx.x < off) red[threadIdx.x] += red[threadIdx.x + off];
    __syncthreads();
  }
  if (threadIdx.x == 0) w[CORR + 256] = red[0];
}

// ---------------------------------------------------------------------------
// Kernel 3: out[s, b, c] = corr / sqrt((win_x2 - win_x^2/T) * yc2)
// ---------------------------------------------------------------------------
__global__ void finalize_kernel(const float* __restrict__ ws,
                                float* __restrict__ out, int T, int total) {
  const int idx = blockIdx.x * blockDim.x + threadIdx.x;
  if (idx >= total) return;
  const int s  = idx >> 3;   // / 8 channels
  const int bc = idx & 7;
  const float* w = ws + (size_t)bc * CH_STRIDE;
  const float wx   = w[WIN_X + s];
  const float wx2  = w[WIN_X2 + s];
  const float num  = w[CORR + s];
  const float xnrm = wx2 - wx * wx / (float)T;
  out[idx] = num * rsqrtf(xnrm * w[YC2]);
}

extern "C" void kernel_launch(void* const* d_in, const int* in_sizes, int n_in,
                              void* d_out, int out_size, void* d_ws, size_t ws_size,
                              hipStream_t stream) {
  const float* x = (const float*)d_in[0];   // (B, C, T) fp32
  const float* y = (const float*)d_in[1];   // (B, C, T) fp32
  float* out = (float*)d_out;               // (S, B, C) fp32
  float* ws  = (float*)d_ws;

  const int channels = 8;                   // B*C = 4*2
  const int T = in_sizes[0] / channels;     // 32768
  // max_shift fixed at 128 (S = 257); the 16x16 shift tiling is built for it.

  stats_kernel<<<channels, 256, 0, stream>>>(x, y, ws, T);
  corr_wmma_kernel<<<channels, 256, 0, stream>>>(x, y, ws, T);

  const int total = NSHIFT * channels;      // 2056
  finalize_kernel<<<(total + 255) / 256, 256, 0, stream>>>(ws, out, T, total);
}